// NSVQ_26826365730937
// MI455X (gfx1250) — compile-verified
//
#include <hip/hip_runtime.h>
#include <math.h>

typedef __attribute__((ext_vector_type(2))) float v2f;
typedef __attribute__((ext_vector_type(4))) float v4f;
typedef __attribute__((ext_vector_type(8))) float v8f;

// Problem constants
#define Dd      64
#define HW      1024
#define Mm      1024
#define ZQ_ELEMS 2097152       // B*D*H*W
#define LOSS_OFF ZQ_ELEMS
#define IND_OFF  (ZQ_ELEMS + 1)
#define LOSS_SCALE (1.0f / 33554432.0f)   // 1/(B*HW*HW)

// Codebook chunking: 64 rows per chunk, double buffered in LDS.
// Rows padded to 68 floats so ds_load_b64 at column 4s+2hi hits banks
// 4r+4s+{0..3} -> all 64 banks, conflict-free.
#define CB_ROWS 64
#define CB_PAD  68
#define NCHUNK  (Mm / CB_ROWS)   // 16

// ---------------------------------------------------------------------------
// Pass 1: y2[m] = sum_d codebook[m,d]^2 ; also zero the loss accumulator.
// ---------------------------------------------------------------------------
__global__ void nsvq_y2_kernel(const float* __restrict__ cb,
                               float* __restrict__ y2,
                               float* __restrict__ lossp) {
    int m = blockIdx.x * blockDim.x + threadIdx.x;
    if (m == 0) *lossp = 0.0f;
    if (m < Mm) {
        const v4f* row = (const v4f*)(cb + m * Dd);
        float s = 0.0f;
#pragma unroll
        for (int i = 0; i < Dd / 4; ++i) {
            v4f v = row[i];
            s += v.x * v.x + v.y * v.y + v.z * v.z + v.w * v.w;
        }
        y2[m] = s;
    }
}

// ---------------------------------------------------------------------------
// Async global->LDS staging of one 64-row codebook chunk (16KB / WG).
// 256 threads x 4 x b128 = 16KB; 4 async ops per wave (uniform ASYNCcnt).
// ---------------------------------------------------------------------------
__device__ __forceinline__ void async_cb_chunk(const float* __restrict__ cb,
                                               float* lds_buf, int chunk, int tid) {
#pragma unroll
    for (int i = 0; i < 4; ++i) {
        const int u   = tid + i * 256;        // 1024 float4 units
        const int row = u >> 4;
        const int c4  = (u & 15) * 4;
        const unsigned goff = (unsigned)(((chunk * CB_ROWS + row) * Dd + c4) * 4);
        const unsigned ldsa = (unsigned)(unsigned long long)(lds_buf + row * CB_PAD + c4);
        asm volatile("global_load_async_to_lds_b128 %0, %1, %2"
                     :: "v"(ldsa), "v"(goff), "s"(cb) : "memory");
    }
}

// ---------------------------------------------------------------------------
// Main kernel: one wave per 16-row tile of zf (16 consecutive n).
// 8 waves / WG; codebook streamed through double-buffered LDS via
// GLOBAL_LOAD_ASYNC_TO_LDS_B128 (ASYNCcnt), overlapped with WMMA compute.
// score[m,n] = dot(zf[n], cb[m]) - 0.5*y2[m]   (argmax == argmin of dist)
// ---------------------------------------------------------------------------
__global__ __launch_bounds__(256) void nsvq_main_kernel(
    const float* __restrict__ z,      // (B,D,H,W)
    const float* __restrict__ cb,     // (M,D)
    const float* __restrict__ noise,  // (N,D)
    const float* __restrict__ y2g,    // (M,)
    float* __restrict__ out) {        // [z_q_out | loss | ind]

    __shared__ alignas(16) float lds_cb[2][CB_ROWS * CB_PAD];
    __shared__ alignas(16) float lds_y2[Mm];

    const int tid  = threadIdx.x;
    const int wave = tid >> 5;
    const int lane = tid & 31;
    const int nl   = lane & 15;   // column within 16-wide tile
    const int hi   = lane >> 4;   // half-wave select

    const int tile  = blockIdx.x * 8 + wave;   // 0..2047
    const int nbase = tile * 16;
    const int bb    = tile >> 6;               // batch index
    const int hw0   = (tile & 63) * 16 + nl;   // this lane's h*W+w

    // ---- Prologue: one-time y2 preload + kick off chunk 0 async copy.
    *(v4f*)(&lds_y2[tid * 4]) = *(const v4f*)(y2g + tid * 4);
    async_cb_chunk(cb, lds_cb[0], 0, tid);

    // ---- Load B operand: zf^T tile (K=64 x N=16), fused transpose from z.
    // Lane(hi): step s holds d = 4s+2hi, 4s+2hi+1 for n = nbase+nl.
    const float* zb = z + (size_t)bb * Dd * HW;
    v2f zfr[16];
#pragma unroll
    for (int s = 0; s < 16; ++s) {
        const int d0 = 4 * s + 2 * hi;
        zfr[s].x = zb[d0 * HW + hw0];
        zfr[s].y = zb[(d0 + 1) * HW + hw0];
    }

    float bestv = -3.402823466e38f;
    int   besti = 0;

    for (int c = 0; c < NCHUNK; ++c) {
        const float* cur = lds_cb[c & 1];
        if (c + 1 < NCHUNK) {
            // Prefetch next chunk into the other buffer, then wait for the
            // current chunk only (async loads complete in order).
            async_cb_chunk(cb, lds_cb[(c + 1) & 1], c + 1, tid);
            asm volatile("s_wait_asynccnt 4" ::: "memory");
        } else {
            asm volatile("s_wait_asynccnt 0" ::: "memory");
        }
        __syncthreads();   // cur chunk visible to all waves

#pragma unroll
        for (int mt = 0; mt < CB_ROWS / 16; ++mt) {
            const int mbase = c * CB_ROWS + mt * 16;
            const int mrow  = (mt * 16 + nl) * CB_PAD;   // A row in LDS

            // acc[r] holds (m_local = r + 8*hi); init with -0.5*y2[m].
            v8f acc;
#pragma unroll
            for (int r = 0; r < 8; ++r)
                acc[r] = -0.5f * lds_y2[mbase + r + 8 * hi];

#pragma unroll
            for (int s = 0; s < 16; ++s) {
                const int d0 = 4 * s + 2 * hi;
                v2f a;
                a.x = cur[mrow + d0];
                a.y = cur[mrow + d0 + 1];
                acc = __builtin_amdgcn_wmma_f32_16x16x4_f32(
                    false, a, false, zfr[s], (short)0, acc, false, false);
            }

            // Running argmax (m ascending; strict > keeps earliest index).
#pragma unroll
            for (int r = 0; r < 8; ++r) {
                const float v = acc[r];
                const int   mi = mbase + r + 8 * hi;
                if (v > bestv) { bestv = v; besti = mi; }
            }
        }
        __syncthreads();   // all waves done reading cur before it is reused
    }

    // ---- Merge the two half-waves (lane L and L^16 share the same n).
    {
        const float bvo = __shfl_xor(bestv, 16);
        const int   bio = __shfl_xor(besti, 16);
        if (bvo > bestv || (bvo == bestv && bio < besti)) { bestv = bvo; besti = bio; }
    }

    // ---- x2[n] = |zf[n]|^2 via half-wave partials.
    float x2p = 0.0f;
#pragma unroll
    for (int s = 0; s < 16; ++s)
        x2p += zfr[s].x * zfr[s].x + zfr[s].y * zfr[s].y;
    const float x2 = x2p + __shfl_xor(x2p, 16);

    // ---- Noise row (same d pattern; contiguous in (N,D) -> float2 loads).
    const float* nrow = noise + (size_t)(nbase + nl) * Dd;
    v2f nz[16];
    float npp = 0.0f;
#pragma unroll
    for (int s = 0; s < 16; ++s) {
        const int d0 = 4 * s + 2 * hi;
        nz[s] = *(const v2f*)(nrow + d0);
        npp += nz[s].x * nz[s].x + nz[s].y * nz[s].y;
    }
    const float nrm   = sqrtf(npp + __shfl_xor(npp, 16));
    const float dmin  = fmaxf(x2 - 2.0f * bestv, 0.0f);
    const float scale = sqrtf(dmin) / fmaxf(nrm, 1e-9f);

    // ---- z_q = zf + noise*scale; store in (B,D,H,W); accumulate tile sum.
    float* ob = out + (size_t)bb * Dd * HW;
    float ts = 0.0f;
#pragma unroll
    for (int s = 0; s < 16; ++s) {
        const int d0 = 4 * s + 2 * hi;
        const float q0 = zfr[s].x + nz[s].x * scale;
        const float q1 = zfr[s].y + nz[s].y * scale;
        ob[d0 * HW + hw0]       = q0;
        ob[(d0 + 1) * HW + hw0] = q1;
        ts += q0 + q1;
    }

    // ---- Tile (b,d)-chunk sum == s[b,d]; wave-reduce then atomic.
#pragma unroll
    for (int off = 1; off < 32; off <<= 1) ts += __shfl_xor(ts, off);
    if (lane == 0)
        atomicAdd(out + LOSS_OFF, ts * ts * LOSS_SCALE);

    // ---- Indices (lanes 0..15 hold final result for n = nbase+lane).
    if (lane < 16)
        out[IND_OFF + nbase + lane] = (float)besti;
}

// ---------------------------------------------------------------------------
extern "C" void kernel_launch(void* const* d_in, const int* in_sizes, int n_in,
                              void* d_out, int out_size, void* d_ws, size_t ws_size,
                              hipStream_t stream) {
    const float* z     = (const float*)d_in[0];
    const float* cb    = (const float*)d_in[1];
    const float* noise = (const float*)d_in[2];
    float* out = (float*)d_out;
    float* y2  = (float*)d_ws;   // 1024 floats of scratch

    nsvq_y2_kernel<<<4, 256, 0, stream>>>(cb, y2, out + LOSS_OFF);
    nsvq_main_kernel<<<256, 256, 0, stream>>>(z, cb, noise, y2, out);
}